// TropicalCausalSelfAttention_24764781429504
// MI455X (gfx1250) — compile-verified
//
#include <hip/hip_runtime.h>
#include <hip/hip_bf16.h>
#include <math.h>

typedef __attribute__((ext_vector_type(2))) float v2f;
typedef __attribute__((ext_vector_type(8))) float v8f;
typedef __attribute__((ext_vector_type(4))) int   v4i;

#define T_SEQ 512
#define C_DIM 1024
#define NH    16
#define NKV   8
#define HD    64
#define HD2   32

// ---------------------------------------------------------------------------
// Async memory->LDS staging (gfx1250 GLOBAL_LOAD_ASYNC_TO_LDS_B128, ASYNCcnt),
// with a synchronous float4 fallback if the toolchain lacks the builtin.
// Builtin signature (from clang diagnostic): (v4i addrspace(1)*, v4i
// addrspace(3)*, imm offset, imm cpol).
// ---------------------------------------------------------------------------
#define GLOBAL_AS __attribute__((address_space(1)))
#define LDS_AS    __attribute__((address_space(3)))

#if __has_builtin(__builtin_amdgcn_global_load_async_to_lds_b128)
#define HAVE_ASYNC_LDS 1
#define ASYNC_COPY16(gp, lp)                                                   \
  __builtin_amdgcn_global_load_async_to_lds_b128(                              \
      (GLOBAL_AS v4i*)(unsigned long long)(gp),                                \
      (LDS_AS v4i*)(unsigned)(unsigned long long)(lp), 0, 0)
#else
#define HAVE_ASYNC_LDS 0
#define ASYNC_COPY16(gp, lp) (*(float4*)(lp) = *(const float4*)(gp))
#endif

#if HAVE_ASYNC_LDS
#if __has_builtin(__builtin_amdgcn_s_wait_asynccnt)
#define WAIT_ASYNC(n) __builtin_amdgcn_s_wait_asynccnt(n)
#else
#define WAIT_ASYNC(n) asm volatile("s_wait_asynccnt %0" ::"i"(n) : "memory")
#endif
#else
#define WAIT_ASYNC(n) ((void)0)
#endif

// ---------------------------------------------------------------------------
// Generic f32 GEMM, C = A @ B, using V_WMMA_F32_16X16X4_F32.
// Block tile 64x64, K-chunk 32, 128 threads = 4 waves in a 2x2 grid; each
// wave owns a 32x32 quadrant = 2x2 WMMA tiles (2 A-frags + 2 B-frags feed
// 4 WMMAs -> 2x fragment reuse). Tiles double-buffered in LDS and staged
// with async LDS-DMA so chunk i+1 copies overlap chunk i compute.
// All launches use M,N,K that are exact multiples of the tile (no guards,
// EXEC stays all-ones as WMMA requires).
// ---------------------------------------------------------------------------
#define BM 64
#define BN 64
#define BK 32
#define SAPAD 36   // A pad: 36*l16 mod 64 distinct over 16 lanes -> no conflicts
#define SBPAD 72   // B pad: koff row pairs land on disjoint bank ranges

__global__ __launch_bounds__(128)
void gemm_f32_wmma(const float* __restrict__ A, const float* __restrict__ B,
                   float* __restrict__ C, int M, int N, int K,
                   int lda, int ldb, int ldc,
                   long long aHeadStride, int bHeadColStride, int bHeadDiv,
                   int cHeadColStride)
{
  const int h = blockIdx.z;
  A += (long long)h * aHeadStride;
  B += (long long)(h / bHeadDiv) * bHeadColStride;
  C += (long long)h * cHeadColStride;

  const int m0   = blockIdx.y * BM;
  const int n0   = blockIdx.x * BN;
  const int tid  = threadIdx.x;
  const int lane = tid & 31;
  const int wave = tid >> 5;
  const int wm   = wave >> 1;         // 2x2 wave grid over the 64x64 tile
  const int wn   = wave & 1;
  const int l16  = lane & 15;
  const int koff = (lane >> 4) * 2;   // lanes 0-15: K=0,1  lanes 16-31: K=2,3

  __shared__ __align__(16) float sA[2][BM][SAPAD];  // A row-major
  __shared__ __align__(16) float sB[2][BK][SBPAD];  // B row-major [k][n]

  v8f acc[2][2];
  #pragma unroll
  for (int mi = 0; mi < 2; ++mi)
    #pragma unroll
    for (int ni = 0; ni < 2; ++ni)
      acc[mi][ni] = (v8f){0.f,0.f,0.f,0.f,0.f,0.f,0.f,0.f};

  // Stage one K-chunk: A 64x32 + B 32x64 = 1024 float4s, 8 per thread.
  auto stage = [&](int buf, int kk) {
    #pragma unroll
    for (int q = 0; q < 4; ++q) {
      int q4 = tid + q * 128;
      {
        int r = q4 >> 3, c4 = q4 & 7;               // A: 8 float4 per row
        const float* gp = A + (long long)(m0 + r) * lda + kk + c4 * 4;
        ASYNC_COPY16(gp, &sA[buf][r][c4 * 4]);
      }
      {
        int r = q4 >> 4, c4 = q4 & 15;              // B: 16 float4 per row
        const float* gp = B + (long long)(kk + r) * ldb + n0 + c4 * 4;
        ASYNC_COPY16(gp, &sB[buf][r][c4 * 4]);
      }
    }
  };

  stage(0, 0);
  int buf = 0;
  for (int kk = 0; kk < K; kk += BK, buf ^= 1) {
    const bool has_next = (kk + BK) < K;
    if (has_next) {
      __builtin_prefetch(&A[(long long)m0 * lda + kk + 2 * BK], 0, 0);
      stage(buf ^ 1, kk + BK);                // overlap DMA with compute
      WAIT_ASYNC(8);                          // current buffer's 8 done
    } else {
      WAIT_ASYNC(0);
    }
    __syncthreads();                          // everyone's staging visible

    const float (*pA)[SAPAD] = sA[buf];
    const float (*pB)[SBPAD] = sB[buf];
    #pragma unroll
    for (int k4 = 0; k4 < BK; k4 += 4) {
      v2f a0 = *(const v2f*)&pA[wm * 32 + l16][k4 + koff];         // ds b64
      v2f a1 = *(const v2f*)&pA[wm * 32 + 16 + l16][k4 + koff];
      v2f b0, b1;                                                  // ds 2addr b32
      b0.x = pB[k4 + koff    ][wn * 32 + l16];
      b0.y = pB[k4 + koff + 1][wn * 32 + l16];
      b1.x = pB[k4 + koff    ][wn * 32 + 16 + l16];
      b1.y = pB[k4 + koff + 1][wn * 32 + 16 + l16];
      // 8 args: (neg_a, A, neg_b, B, c_mod, C, reuse_a, reuse_b)
      acc[0][0] = __builtin_amdgcn_wmma_f32_16x16x4_f32(false, a0, false, b0,
                                                        (short)0, acc[0][0], false, false);
      acc[0][1] = __builtin_amdgcn_wmma_f32_16x16x4_f32(false, a0, false, b1,
                                                        (short)0, acc[0][1], false, false);
      acc[1][0] = __builtin_amdgcn_wmma_f32_16x16x4_f32(false, a1, false, b0,
                                                        (short)0, acc[1][0], false, false);
      acc[1][1] = __builtin_amdgcn_wmma_f32_16x16x4_f32(false, a1, false, b1,
                                                        (short)0, acc[1][1], false, false);
    }
    __syncthreads();                          // done reading before restage
  }

  // C/D layout: VGPR r -> M = r (lanes 0-15) / M = 8+r (lanes 16-31), N = lane%16.
  #pragma unroll
  for (int mi = 0; mi < 2; ++mi)
    #pragma unroll
    for (int ni = 0; ni < 2; ++ni) {
      int col  = n0 + wn * 32 + ni * 16 + l16;
      int row0 = m0 + wm * 32 + mi * 16 + (lane >> 4) * 8;
      #pragma unroll
      for (int r = 0; r < 8; ++r)
        C[(long long)(row0 + r) * ldc + col] = acc[mi][ni][r];
    }
}

// ---------------------------------------------------------------------------
// RoPE + per-head RMS-norm, in place. One wave32 per (token, head); lane l
// owns dims l and l+32 of the 64-dim head.
// ---------------------------------------------------------------------------
__global__ __launch_bounds__(256)
void rope_rmsnorm(float* __restrict__ t, const float* __restrict__ cosp,
                  const float* __restrict__ sinp, int nHeads)
{
  int wid  = blockIdx.x * (blockDim.x >> 5) + (threadIdx.x >> 5);
  int lane = threadIdx.x & 31;
  int tpos = wid / nHeads;
  int head = wid - tpos * nHeads;
  if (tpos >= T_SEQ) return;
  float* p = t + (long long)tpos * nHeads * HD + head * HD;

  float x1 = p[lane], x2 = p[lane + HD2];
  float c  = cosp[tpos * HD2 + lane];
  float s  = sinp[tpos * HD2 + lane];
  float r1 = x1 * c - x2 * s;
  float r2 = x1 * s + x2 * c;

  float ss = r1 * r1 + r2 * r2;
  #pragma unroll
  for (int off = 16; off > 0; off >>= 1) ss += __shfl_xor(ss, off, 32);
  float scale = rsqrtf(ss * (1.0f / HD) + 1e-6f);
  p[lane]       = r1 * scale;
  p[lane + HD2] = r2 * scale;
}

// ---------------------------------------------------------------------------
// Tropical (max-plus) scores: S[h,i,j] = max_d(q[i,hd+d] + k[j,kvd+d]).
// 32x32 tile per block; q rows broadcast from LDS, k transposed in LDS for
// conflict-free per-lane key reads. Blocks fully above the diagonal skipped;
// the causal softmax never reads j > i, so garbage there is harmless.
// ---------------------------------------------------------------------------
__global__ __launch_bounds__(256)
void tropical_scores(const float* __restrict__ q, const float* __restrict__ k,
                     float* __restrict__ S)
{
  int h  = blockIdx.z;
  int i0 = blockIdx.y * 32, j0 = blockIdx.x * 32;
  if (j0 > i0 + 31) return;   // uniform over the whole block

  __shared__ __align__(16) float qs[32][68];
  __shared__ float ksT[HD][33];
  int tid = threadIdx.x;

  #pragma unroll
  for (int t = 0; t < 2; ++t) {
    int q4 = tid + t * 256;     // 512 float4s for each of q / k tiles
    int r  = q4 >> 4, c4 = q4 & 15;
    float4 v = *(const float4*)(q + (long long)(i0 + r) * C_DIM + h * HD + c4 * 4);
    *(float4*)&qs[r][c4 * 4] = v;
    float4 kv = *(const float4*)(k + (long long)(j0 + r) * (NKV * HD)
                                 + (h >> 1) * HD + c4 * 4);
    ksT[c4*4 + 0][r] = kv.x; ksT[c4*4 + 1][r] = kv.y;
    ksT[c4*4 + 2][r] = kv.z; ksT[c4*4 + 3][r] = kv.w;
  }
  __syncthreads();

  int ti = tid >> 5, tj = tid & 31;   // wave owns 4 i-rows x 32 j-cols
  float sc[4] = {-3.0e38f, -3.0e38f, -3.0e38f, -3.0e38f};
  #pragma unroll 8
  for (int d = 0; d < HD; ++d) {
    float kv = ksT[d][tj];
    #pragma unroll
    for (int r = 0; r < 4; ++r)
      sc[r] = fmaxf(sc[r], qs[ti * 4 + r][d] + kv);
  }
  long long base = (long long)h * T_SEQ * T_SEQ + (long long)i0 * T_SEQ + j0 + tj;
  #pragma unroll
  for (int r = 0; r < 4; ++r)
    S[base + (long long)(ti * 4 + r) * T_SEQ] = sc[r];
}

// ---------------------------------------------------------------------------
// Causal softmax, in place. One wave32 per (head, row); 16 cols per lane.
// Writes exact zeros for j > i so the P@V GEMM can run the full K range.
// ---------------------------------------------------------------------------
__global__ __launch_bounds__(256)
void softmax_causal(float* __restrict__ P)
{
  int wid  = blockIdx.x * (blockDim.x >> 5) + (threadIdx.x >> 5);
  int lane = threadIdx.x & 31;
  int h = wid >> 9;
  int i = wid & (T_SEQ - 1);
  if (h >= NH) return;
  float* row = P + (long long)h * T_SEQ * T_SEQ + (long long)i * T_SEQ;

  float v[16];
  float m = -3.0e38f;
  #pragma unroll
  for (int jj = 0; jj < 16; ++jj) {
    int j = lane + jj * 32;
    float val = (j <= i) ? row[j] : -3.0e38f;
    v[jj] = val;
    m = fmaxf(m, val);
  }
  #pragma unroll
  for (int off = 16; off > 0; off >>= 1) m = fmaxf(m, __shfl_xor(m, off, 32));

  float sum = 0.f;
  #pragma unroll
  for (int jj = 0; jj < 16; ++jj) {
    int j = lane + jj * 32;
    float e = (j <= i) ? __expf(v[jj] - m) : 0.f;
    v[jj] = e;
    sum += e;
  }
  #pragma unroll
  for (int off = 16; off > 0; off >>= 1) sum += __shfl_xor(sum, off, 32);

  float inv = 1.0f / sum;   // row i always has valid j=0 -> sum >= exp(0) > 0
  #pragma unroll
  for (int jj = 0; jj < 16; ++jj) row[lane + jj * 32] = v[jj] * inv;
}

// ---------------------------------------------------------------------------
extern "C" void kernel_launch(void* const* d_in, const int* in_sizes, int n_in,
                              void* d_out, int out_size, void* d_ws, size_t ws_size,
                              hipStream_t stream) {
  const float* x    = (const float*)d_in[0];
  const float* cosp = (const float*)d_in[1];
  const float* sinp = (const float*)d_in[2];
  const float* Wq   = (const float*)d_in[3];
  const float* Wk   = (const float*)d_in[4];
  const float* Wv   = (const float*)d_in[5];
  const float* Wo   = (const float*)d_in[6];
  float*       out  = (float*)d_out;

  float* ws   = (float*)d_ws;
  float* qbuf = ws;                                   // 512*1024
  float* kbuf = qbuf + (size_t)T_SEQ * C_DIM;         // 512*512
  float* vbuf = kbuf + (size_t)T_SEQ * NKV * HD;      // 512*512
  float* sbuf = vbuf + (size_t)T_SEQ * NKV * HD;      // 16*512*512
  float* ybuf = sbuf + (size_t)NH * T_SEQ * T_SEQ;    // 512*1024

  dim3 blk(128);

  // q = x @ Wq   (512x1024x1024)
  gemm_f32_wmma<<<dim3(C_DIM / BN, T_SEQ / BM, 1), blk, 0, stream>>>(
      x, Wq, qbuf, T_SEQ, C_DIM, C_DIM, C_DIM, C_DIM, C_DIM, 0LL, 0, 1, 0);
  // k = x @ Wk, v = x @ Wv   (512x512x1024)
  gemm_f32_wmma<<<dim3((NKV * HD) / BN, T_SEQ / BM, 1), blk, 0, stream>>>(
      x, Wk, kbuf, T_SEQ, NKV * HD, C_DIM, C_DIM, NKV * HD, NKV * HD, 0LL, 0, 1, 0);
  gemm_f32_wmma<<<dim3((NKV * HD) / BN, T_SEQ / BM, 1), blk, 0, stream>>>(
      x, Wv, vbuf, T_SEQ, NKV * HD, C_DIM, C_DIM, NKV * HD, NKV * HD, 0LL, 0, 1, 0);

  // RoPE + RMS-norm on q (16 heads) and k (8 heads), in place.
  rope_rmsnorm<<<dim3((T_SEQ * NH) / 8), dim3(256), 0, stream>>>(qbuf, cosp, sinp, NH);
  rope_rmsnorm<<<dim3((T_SEQ * NKV) / 8), dim3(256), 0, stream>>>(kbuf, cosp, sinp, NKV);

  // Tropical max-plus scores (VALU) + causal softmax.
  tropical_scores<<<dim3(T_SEQ / 32, T_SEQ / 32, NH), dim3(256), 0, stream>>>(
      qbuf, kbuf, sbuf);
  softmax_causal<<<dim3((NH * T_SEQ) / 8), dim3(256), 0, stream>>>(sbuf);

  // y[h] = P[h] @ V[h/2]   (per head 512x64x512, GQA via bHeadDiv=2)
  gemm_f32_wmma<<<dim3(HD / BN, T_SEQ / BM, NH), blk, 0, stream>>>(
      sbuf, vbuf, ybuf, T_SEQ, HD, T_SEQ,
      T_SEQ, NKV * HD, C_DIM,
      (long long)T_SEQ * T_SEQ, HD, 2, HD);

  // out = y @ Wo   (512x1024x1024)
  gemm_f32_wmma<<<dim3(C_DIM / BN, T_SEQ / BM, 1), blk, 0, stream>>>(
      ybuf, Wo, out, T_SEQ, C_DIM, C_DIM, C_DIM, C_DIM, C_DIM, 0LL, 0, 1, 0);
}